// NCC_23630910062853
// MI455X (gfx1250) — compile-verified
//
#include <hip/hip_runtime.h>
#include <hip/hip_bf16.h>

// NCC (local normalized cross-correlation), 9^3 window, (2,1,128,128,128) f32.
// Separable box sums: pass1 z (scalar sliding window), pass2 y (scalar sliding
// window), pass3 x via V_WMMA_F32_16X16X4_F32 ones-band matmul + cc +
// deterministic reduction. Pass-3 LDS tile carries a 4-column zero halo so all
// B-operand fetches are unconditional 8B ds_load_b64 (no EXEC-masked loads).

typedef __attribute__((ext_vector_type(2))) float v2f;
typedef __attribute__((ext_vector_type(8))) float v8f;

#define NVOX 4194304      // 2*1*128*128*128
#define DIM  128
#define WINR 4            // radius
#define WSIZ 729.0f       // 9^3
#define NCCEPS 1e-6f
#define NLINES 32768      // 2*128*128 lines per axis
#define LDSROW 140        // 4 halo + 128 + 4 halo, padded: n*140%64 distinct -> no bank conflicts

// ---------------- Pass 1: box-sum along z, build 5 channels ----------------
__global__ __launch_bounds__(256) void ncc_pass1_zsum(const float* __restrict__ I,
                                                      const float* __restrict__ J,
                                                      float* __restrict__ A) {
    int t = blockIdx.x * 256 + threadIdx.x;      // one thread per (b,y,x) line
    int b  = t >> 14;
    int yx = t & 16383;
    size_t base = ((size_t)b << 21) + (size_t)yx;   // idx = b*2^21 + z*2^14 + yx
    float s0 = 0.f, s1 = 0.f, s2 = 0.f, s3 = 0.f, s4 = 0.f;
    for (int zl = 0; zl < DIM + WINR; ++zl) {
        if (zl < DIM) {
            size_t idx = base + ((size_t)zl << 14);
            float i = I[idx], j = J[idx];
            s0 += i; s1 += j; s2 += i * i; s3 += j * j; s4 += i * j;
        }
        int zs = zl - (2 * WINR + 1);
        if (zs >= 0) {
            size_t idx = base + ((size_t)zs << 14);
            float i = I[idx], j = J[idx];
            s0 -= i; s1 -= j; s2 -= i * i; s3 -= j * j; s4 -= i * j;
        }
        int zo = zl - WINR;
        if (zo >= 0 && zo < DIM) {
            size_t o = base + ((size_t)zo << 14);
            A[o]                   = s0;
            A[o + (size_t)NVOX]    = s1;
            A[o + (size_t)2*NVOX]  = s2;
            A[o + (size_t)3*NVOX]  = s3;
            A[o + (size_t)4*NVOX]  = s4;
        }
    }
}

// ---------------- Pass 2: box-sum along y on the 5 channels ----------------
__global__ __launch_bounds__(256) void ncc_pass2_ysum(const float* __restrict__ A,
                                                      float* __restrict__ B) {
    int t = blockIdx.x * 256 + threadIdx.x;      // one thread per (b,z,x) line
    int x = t & 127;
    int z = (t >> 7) & 127;
    int b = t >> 14;
    size_t base = ((size_t)b << 21) + ((size_t)z << 14) + (size_t)x; // + y*128
    float s[5] = {0.f, 0.f, 0.f, 0.f, 0.f};
    for (int yl = 0; yl < DIM + WINR; ++yl) {
        if (yl < DIM) {
            size_t idx = base + ((size_t)yl << 7);
#pragma unroll
            for (int c = 0; c < 5; ++c) s[c] += A[(size_t)c * NVOX + idx];
        }
        int ys = yl - (2 * WINR + 1);
        if (ys >= 0) {
            size_t idx = base + ((size_t)ys << 7);
#pragma unroll
            for (int c = 0; c < 5; ++c) s[c] -= A[(size_t)c * NVOX + idx];
        }
        int yo = yl - WINR;
        if (yo >= 0 && yo < DIM) {
            size_t o = base + ((size_t)yo << 7);
#pragma unroll
            for (int c = 0; c < 5; ++c) B[(size_t)c * NVOX + o] = s[c];
        }
    }
}

// ---- Pass 3: box-sum along x as a ones-band matmul (WMMA f32 16x16x4),
//      then cc formula + block reduction. One block = 16 contiguous lines. ----
__global__ __launch_bounds__(128) void ncc_pass3_wmma(const float* __restrict__ B,
                                                      float* __restrict__ partial) {
    __shared__ float tile[5][16][LDSROW];   // [ch][line][4 halo | 128 | 4 halo]
    __shared__ float red[128];
    const int tid = threadIdx.x;                 // 128 threads = 4 waves
    const size_t base = (size_t)blockIdx.x * 2048;   // 16 lines * 128 x, contiguous

    // Zero the 4-column halos on both sides of each of the 5*16 rows (640 elems).
    for (int i = tid; i < 640; i += 128) {
        int c   = i >> 7;            // /128
        int rem = i & 127;
        int row = rem >> 3;
        int h   = rem & 7;
        int x   = (h < 4) ? h : (128 + h);   // 0..3 and 132..135
        tile[c][row][x] = 0.f;
    }
    // Stage 16 lines x 128 x, 5 channels, coalesced global -> padded LDS rows.
#pragma unroll
    for (int c = 0; c < 5; ++c) {
        for (int i = tid; i < 2048; i += 128) {
            tile[c][i >> 7][(i & 127) + WINR] = B[(size_t)c * NVOX + base + i];
        }
    }
    __syncthreads();

    const int lane = tid & 31;
    const int wave = tid >> 5;
    const int n  = lane & 15;     // N index (line)
    const int hi = lane >> 4;     // half-wave selector in A/B K layout

    float acc = 0.f;
    // Each wave does two 16-wide x-chunks: x0 = 16*(2*wave), 16*(2*wave+1).
    for (int chk = 0; chk < 2; ++chk) {
        const int x0 = (wave * 2 + chk) * 16;
        v8f d0 = {}, d1 = {}, d2 = {}, d3 = {}, d4 = {};
#pragma unroll
        for (int cc = 0; cc < 6; ++cc) {          // K_total = 24 = 6 chunks of 4
            const int kk0 = cc * 4 + 2 * hi;      // A layout: lane holds K=2*hi+{0,1}
            // Ones-band: A[m,kk] = 1 iff m <= kk <= m+8  (m = n = lane&15)
            v2f a;
            a.x = (kk0     >= n && kk0     <= n + 8) ? 1.f : 0.f;
            a.y = (kk0 + 1 >= n && kk0 + 1 <= n + 8) ? 1.f : 0.f;
            // B element x-input = x0 - 4 + kk; halo offset +4 => LDS col = x0 + kk0.
            // Even col, even row stride -> 8-byte aligned ds_load_b64.
            const int col = x0 + kk0;
            v2f bv;
            bv = *(const v2f*)&tile[0][n][col];
            d0 = __builtin_amdgcn_wmma_f32_16x16x4_f32(false, a, false, bv, (short)0, d0, false, false);
            bv = *(const v2f*)&tile[1][n][col];
            d1 = __builtin_amdgcn_wmma_f32_16x16x4_f32(false, a, false, bv, (short)0, d1, false, false);
            bv = *(const v2f*)&tile[2][n][col];
            d2 = __builtin_amdgcn_wmma_f32_16x16x4_f32(false, a, false, bv, (short)0, d2, false, false);
            bv = *(const v2f*)&tile[3][n][col];
            d3 = __builtin_amdgcn_wmma_f32_16x16x4_f32(false, a, false, bv, (short)0, d3, false, false);
            bv = *(const v2f*)&tile[4][n][col];
            d4 = __builtin_amdgcn_wmma_f32_16x16x4_f32(false, a, false, bv, (short)0, d4, false, false);
        }
        // D layout: VGPR r, lanes 0-15 -> M=r, lanes 16-31 -> M=r+8; each lane
        // owns 8 unique voxels (x = x0 + r + 8*hi, line n). Apply cc formula.
        const float invW = 1.0f / WSIZ;
#pragma unroll
        for (int r = 0; r < 8; ++r) {
            float Is = d0[r], Js = d1[r], I2 = d2[r], J2 = d3[r], IJ = d4[r];
            float cross = IJ - Is * Js * invW;
            float Ivar  = I2 - Is * Is * invW;
            float Jvar  = J2 - Js * Js * invW;
            float den = sqrtf(fmaxf(Ivar, NCCEPS)) * sqrtf(fmaxf(Jvar, NCCEPS));
            acc += cross / den;
        }
    }

    // Deterministic block reduction.
    red[tid] = acc;
    __syncthreads();
#pragma unroll
    for (int sft = 64; sft > 0; sft >>= 1) {
        if (tid < sft) red[tid] += red[tid + sft];
        __syncthreads();
    }
    if (tid == 0) partial[blockIdx.x] = red[0];
}

// ---------------- Final: sum 2048 partials, emit -mean(cc)+1 ----------------
__global__ __launch_bounds__(256) void ncc_final(const float* __restrict__ partial,
                                                 float* __restrict__ out) {
    __shared__ float red[256];
    int tid = threadIdx.x;
    float s = 0.f;
    for (int i = tid; i < 2048; i += 256) s += partial[i];
    red[tid] = s;
    __syncthreads();
#pragma unroll
    for (int k = 128; k > 0; k >>= 1) {
        if (tid < k) red[tid] += red[tid + k];
        __syncthreads();
    }
    if (tid == 0) out[0] = 1.0f - red[0] * (1.0f / (float)NVOX);
}

extern "C" void kernel_launch(void* const* d_in, const int* in_sizes, int n_in,
                              void* d_out, int out_size, void* d_ws, size_t ws_size,
                              hipStream_t stream) {
    (void)in_sizes; (void)n_in; (void)out_size; (void)ws_size;
    const float* I = (const float*)d_in[0];   // y_true
    const float* J = (const float*)d_in[1];   // y_pred
    float* out = (float*)d_out;

    float* A       = (float*)d_ws;                    // 5*NVOX floats
    float* Bws     = A + (size_t)5 * NVOX;            // 5*NVOX floats
    float* partial = Bws + (size_t)5 * NVOX;          // 2048 floats

    ncc_pass1_zsum<<<NLINES / 256, 256, 0, stream>>>(I, J, A);
    ncc_pass2_ysum<<<NLINES / 256, 256, 0, stream>>>(A, Bws);
    ncc_pass3_wmma<<<NLINES / 16, 128, 0, stream>>>(Bws, partial);
    ncc_final<<<1, 256, 0, stream>>>(partial, out);
}